// Grelen_9328668967409
// MI455X (gfx1250) — compile-verified
//
#include <hip/hip_runtime.h>
#include <hip/hip_bf16.h>
#include <math.h>

// ---------------------------------------------------------------------------
// Problem constants (from the reference)
// ---------------------------------------------------------------------------
constexpr int NB    = 16;    // batch
constexpr int NODES = 1024;  // N
constexpr int UU    = 64;    // units
constexpr int FEAT  = 128;   // 2U
constexpr int ROWS  = NB * NODES;  // 16384 flattened rows

typedef __attribute__((ext_vector_type(2))) float v2f;
typedef __attribute__((ext_vector_type(8))) float v8f;

// D(16x16,f32) = A(16x4,f32) x B(4x16,f32) + C  -- CDNA5 WMMA, wave32
__device__ __forceinline__ v8f wmma4(v2f a, v2f b, v8f c) {
  return __builtin_amdgcn_wmma_f32_16x16x4_f32(
      /*neg_a=*/false, a, /*neg_b=*/false, b,
      /*c_mod=*/(short)0, c, /*reuse_a=*/false, /*reuse_b=*/false);
}

// --- CDNA5 async global->LDS copies (ASYNCcnt tracked, no VGPR staging) ----
__device__ __forceinline__ unsigned lds_off(const void* p) {
  // generic LDS pointer = {SHARED_BASE[63:32], lds_offset[31:0]}
  return (unsigned)(unsigned long long)p;
}
__device__ __forceinline__ void async_g2l_b128(unsigned lds, const float* g) {
  asm volatile("global_load_async_to_lds_b128 %0, %1, off"
               :: "v"(lds), "v"(g) : "memory");
}
__device__ __forceinline__ void async_g2l_b32(unsigned lds, const float* g) {
  asm volatile("global_load_async_to_lds_b32 %0, %1, off"
               :: "v"(lds), "v"(g) : "memory");
}
__device__ __forceinline__ void wait_async0() {
  asm volatile("s_wait_asynccnt 0x0" ::: "memory");
}

// Interleave two k-rows (a = row 2kp, b = row 2kp+1) for 4 columns into LDS.
__device__ __forceinline__ void st_pair(float* dst, float4 a, float4 b) {
  float4 lo = make_float4(a.x, b.x, a.y, b.y);
  float4 hi = make_float4(a.z, b.z, a.w, b.w);
  *(float4*)(dst)     = lo;
  *(float4*)(dst + 4) = hi;
}

// ---------------------------------------------------------------------------
// Degree reductions: d0inv = 1/(1+colsum(adj)), d1inv = 1/(1+rowsum(adj))
// ---------------------------------------------------------------------------
__global__ __launch_bounds__(256) void k_deg_col(const float* __restrict__ adj,
                                                 float* __restrict__ d0inv) {
  const int b = blockIdx.y;
  const int i = blockIdx.x * 256 + threadIdx.x;  // column
  const float* A = adj + (size_t)b * NODES * NODES;
  float s = 1.0f;
  for (int k = 0; k < NODES; ++k) s += A[(size_t)k * NODES + i];
  d0inv[b * NODES + i] = (s == 0.0f) ? 0.0f : 1.0f / s;
}

__global__ __launch_bounds__(256) void k_deg_row(const float* __restrict__ adj,
                                                 float* __restrict__ d1inv) {
  const int b    = blockIdx.y;
  const int wave = threadIdx.x >> 5;
  const int lane = threadIdx.x & 31;
  const int i    = blockIdx.x * 8 + wave;  // row
  const float* row = adj + ((size_t)b * NODES + i) * NODES;
  float s = 0.0f;
  for (int k = lane; k < NODES; k += 32) s += row[k];
  #pragma unroll
  for (int off = 16; off > 0; off >>= 1) s += __shfl_xor(s, off, 32);
  if (lane == 0) {
    s += 1.0f;
    d1inv[b * NODES + i] = (s == 0.0f) ? 0.0f : 1.0f / s;
  }
}

// ---------------------------------------------------------------------------
// Weight combining: fold the [x, Ax, Ax] diffusion stack into 3 matrices.
// ---------------------------------------------------------------------------
__global__ __launch_bounds__(128) void k_wcomb(
    const float* __restrict__ W0, const float* __restrict__ W1,
    const float* __restrict__ b0, const float* __restrict__ b1,
    const float* __restrict__ Wc0, const float* __restrict__ Wc1,
    const float* __restrict__ bc0, const float* __restrict__ bc1,
    float* __restrict__ Wx, float* __restrict__ Wy0, float* __restrict__ Wy1,
    float* __restrict__ Wcx, float* __restrict__ Wcy0, float* __restrict__ Wcy1,
    float* __restrict__ bsum, float* __restrict__ bcsum) {
  const int f = blockIdx.x;   // 0..127
  const int o = threadIdx.x;  // 0..127
  Wx [f * 128 + o] = W0[(3 * f    ) * 128 + o] + W1[(3 * f    ) * 128 + o];
  Wy0[f * 128 + o] = W0[(3 * f + 1) * 128 + o] + W0[(3 * f + 2) * 128 + o];
  Wy1[f * 128 + o] = W1[(3 * f + 1) * 128 + o] + W1[(3 * f + 2) * 128 + o];
  if (o < UU) {
    Wcx [f * UU + o] = Wc0[(3 * f    ) * UU + o] + Wc1[(3 * f    ) * UU + o];
    Wcy0[f * UU + o] = Wc0[(3 * f + 1) * UU + o] + Wc0[(3 * f + 2) * UU + o];
    Wcy1[f * UU + o] = Wc1[(3 * f + 1) * UU + o] + Wc1[(3 * f + 2) * UU + o];
  }
  if (f == 0) {
    bsum[o] = b0[o] + b1[o];
    if (o < UU) bcsum[o] = bc0[o] + bc1[o];
  }
}

// ---------------------------------------------------------------------------
// Diffusion GEMM:  y = dinv * (x + adj(^T) @ x)   (per batch)
//   blockIdx = (row_tile, batch, direction)   block = 256 thr = 8 waves
//   A-tile staged via async global->LDS; X-tile stored k-paired for
//   single-ds_load_b64 WMMA B operands.
// ---------------------------------------------------------------------------
template <int FC>
__global__ __launch_bounds__(256) void k_diffuse(
    const float* __restrict__ adj,
    const float* __restrict__ xa,  // features f<64, layout [b*N + n]*64 + f
    const float* __restrict__ xb,  // features f>=64 (FC==128 only)
    const float* __restrict__ d0inv, const float* __restrict__ d1inv,
    float* __restrict__ y0, float* __restrict__ y1) {
  constexpr int NT  = FC / 32;        // 16-col subtiles per wave
  constexpr int XS  = 2 * FC + 32;    // paired-row stride (== 32 mod 64 dwords)
  __shared__ float At[64][68];
  __shared__ float Xp[32][XS];

  const int b    = blockIdx.y;
  const int row0 = blockIdx.x * 64;
  const int dir  = blockIdx.z;
  const int t    = threadIdx.x;
  const int w    = t >> 5;
  const int lane = t & 31;
  const int row16 = lane & 15;
  const int hk    = lane >> 4;
  const int wr = w >> 1;              // 0..3 -> row subtile
  const int wc = w & 1;               // 0..1 -> col half
  const int colbase = wc * (FC / 2);
  const size_t adjB = (size_t)b * NODES * NODES;

  v8f acc[NT];
  #pragma unroll
  for (int j = 0; j < NT; ++j)
    #pragma unroll
    for (int e = 0; e < 8; ++e) acc[j][e] = 0.0f;

  for (int k0 = 0; k0 < NODES; k0 += 64) {
    // ---- A tile: async copy straight into LDS ----
    if (dir == 0) {  // At[r][k] = adj[b][row0+r][k0+k]
      #pragma unroll
      for (int p = 0; p < 4; ++p) {
        int r  = (t >> 4) + p * 16;
        int c4 = (t & 15) * 4;
        async_g2l_b128(lds_off(&At[r][c4]),
                       adj + adjB + (size_t)(row0 + r) * NODES + k0 + c4);
      }
    } else {         // At[r][k] = adj[b][k0+k][row0+r]   (transposed)
      #pragma unroll
      for (int p = 0; p < 16; ++p) {
        int k = (t >> 6) + p * 4;
        int r = t & 63;
        async_g2l_b32(lds_off(&At[r][k]),
                      adj + adjB + (size_t)(k0 + k) * NODES + row0 + r);
      }
    }
    // ---- X tile: k-paired layout Xp[k>>1][2*c + (k&1)] ----
    if (FC == 128) {
      #pragma unroll
      for (int p = 0; p < 4; ++p) {
        int kp = (t >> 5) + p * 8;        // pair index 0..31
        int c  = (t & 31) * 4;
        const int k = k0 + 2 * kp;
        const float* s0;
        const float* s1;
        if (c < 64) {
          s0 = xa + (size_t)(b * NODES + k    ) * 64 + c;
          s1 = xa + (size_t)(b * NODES + k + 1) * 64 + c;
        } else {
          s0 = xb + (size_t)(b * NODES + k    ) * 64 + (c - 64);
          s1 = xb + (size_t)(b * NODES + k + 1) * 64 + (c - 64);
        }
        st_pair(&Xp[kp][2 * c], *(const float4*)s0, *(const float4*)s1);
      }
    } else {
      #pragma unroll
      for (int p = 0; p < 2; ++p) {
        int kp = (t >> 4) + p * 16;
        int c  = (t & 15) * 4;
        const int k = k0 + 2 * kp;
        const float* s0 = xa + (size_t)(b * NODES + k    ) * 64 + c;
        const float* s1 = xa + (size_t)(b * NODES + k + 1) * 64 + c;
        st_pair(&Xp[kp][2 * c], *(const float4*)s0, *(const float4*)s1);
      }
    }
    wait_async0();
    __syncthreads();
    #pragma unroll
    for (int kk = 0; kk < 64; kk += 4) {
      v2f a = *(const v2f*)&At[16 * wr + row16][kk + 2 * hk];
      const int kp = (kk >> 1) + hk;
      #pragma unroll
      for (int j = 0; j < NT; ++j) {
        v2f bv = *(const v2f*)&Xp[kp][2 * (colbase + 16 * j + row16)];
        acc[j] = wmma4(a, bv, acc[j]);
      }
    }
    __syncthreads();
  }

  const float* dinv = dir ? d1inv : d0inv;
  float* yout       = dir ? y1 : y0;
  #pragma unroll
  for (int j = 0; j < NT; ++j) {
    const int col = colbase + 16 * j + row16;
    #pragma unroll
    for (int r = 0; r < 8; ++r) {
      const int gi = row0 + 16 * wr + r + 8 * hk;
      float xv;
      if (FC == 128)
        xv = (col < 64) ? xa[(size_t)(b * NODES + gi) * 64 + col]
                        : xb[(size_t)(b * NODES + gi) * 64 + col - 64];
      else
        xv = xa[(size_t)(b * NODES + gi) * 64 + col];
      yout[((size_t)b * NODES + gi) * FC + col] =
          dinv[b * NODES + gi] * (xv + acc[j][r]);
    }
  }
}

// ---------------------------------------------------------------------------
// Gate GEMM: pre = x@Wx + y0@Wy0 + y1@Wy1 + bsum ; sigmoid ; split into r,u
//   rows = 16384, cols = 128, K = 3 x 128. Block: 64 rows x 128 cols.
// ---------------------------------------------------------------------------
__global__ __launch_bounds__(256) void k_gate(
    const float* __restrict__ inputs, const float* __restrict__ hx,
    const float* __restrict__ y0, const float* __restrict__ y1,
    const float* __restrict__ Wx, const float* __restrict__ Wy0,
    const float* __restrict__ Wy1, const float* __restrict__ bsum,
    float* __restrict__ rg, float* __restrict__ ug) {
  constexpr int BS = 2 * 128 + 32;    // 288
  __shared__ float At[64][68];
  __shared__ float Bp[32][BS];
  const int row0 = blockIdx.x * 64;
  const int t    = threadIdx.x;
  const int w    = t >> 5;
  const int lane = t & 31;
  const int row16 = lane & 15;
  const int hk    = lane >> 4;
  const int wr = w >> 1;
  const int wc = w & 1;
  const int colbase = wc * 64;

  v8f acc[4];
  #pragma unroll
  for (int j = 0; j < 4; ++j)
    #pragma unroll
    for (int e = 0; e < 8; ++e) acc[j][e] = 0.0f;

  for (int src = 0; src < 3; ++src) {
    const float* Wsrc = (src == 0) ? Wx : (src == 1) ? Wy0 : Wy1;
    for (int k0 = 0; k0 < FEAT; k0 += 64) {
      const float* Aptr; size_t stride; int off;
      if (src == 0)      { Aptr = (k0 == 0) ? inputs : hx; stride = 64;  off = 0;  }
      else if (src == 1) { Aptr = y0;                      stride = 128; off = k0; }
      else               { Aptr = y1;                      stride = 128; off = k0; }
      #pragma unroll
      for (int p = 0; p < 4; ++p) {
        int r  = (t >> 4) + p * 16;
        int c4 = (t & 15) * 4;
        float4 v = *(const float4*)(Aptr + (size_t)(row0 + r) * stride + off + c4);
        At[r][c4 + 0] = v.x; At[r][c4 + 1] = v.y;
        At[r][c4 + 2] = v.z; At[r][c4 + 3] = v.w;
      }
      // weights, k-paired: Bp[kp][2*o + (k&1)] = W[(k0+2kp+(k&1))*128 + o]
      #pragma unroll
      for (int p = 0; p < 4; ++p) {
        int kp = (t >> 5) + p * 8;
        int o  = (t & 31) * 4;
        const float* s0 = Wsrc + (size_t)(k0 + 2 * kp    ) * 128 + o;
        const float* s1 = Wsrc + (size_t)(k0 + 2 * kp + 1) * 128 + o;
        st_pair(&Bp[kp][2 * o], *(const float4*)s0, *(const float4*)s1);
      }
      __syncthreads();
      #pragma unroll
      for (int kk = 0; kk < 64; kk += 4) {
        v2f a = *(const v2f*)&At[16 * wr + row16][kk + 2 * hk];
        const int kp = (kk >> 1) + hk;
        #pragma unroll
        for (int j = 0; j < 4; ++j) {
          v2f bv = *(const v2f*)&Bp[kp][2 * (colbase + 16 * j + row16)];
          acc[j] = wmma4(a, bv, acc[j]);
        }
      }
      __syncthreads();
    }
  }

  #pragma unroll
  for (int j = 0; j < 4; ++j) {
    const int col = colbase + 16 * j + row16;
    #pragma unroll
    for (int r = 0; r < 8; ++r) {
      const int gr = row0 + 16 * wr + r + 8 * hk;
      float v = acc[j][r] + bsum[col];
      float g = 1.0f / (1.0f + expf(-v));
      if (col < UU) rg[(size_t)gr * UU + col]      = g;
      else          ug[(size_t)gr * UU + col - UU] = g;
    }
  }
}

// ---------------------------------------------------------------------------
// s = r * hx   (elementwise)
// ---------------------------------------------------------------------------
__global__ __launch_bounds__(256) void k_rh(const float* __restrict__ rg,
                                            const float* __restrict__ hx,
                                            float* __restrict__ s, int n) {
  int i = blockIdx.x * 256 + threadIdx.x;
  if (i < n) s[i] = rg[i] * hx[i];
}

// ---------------------------------------------------------------------------
// Candidate GEMM + GRU blend:
//   c = tanh(xc@Wcx + yc0@Wcy0 + yc1@Wcy1 + bcsum);  out = u*hx + (1-u)*c
//   xc = [inputs|s]; yc0 = [y0[:, :64]|yc0s]; yc1 = [y1[:, :64]|yc1s]
// ---------------------------------------------------------------------------
__global__ __launch_bounds__(256) void k_cand(
    const float* __restrict__ inputs, const float* __restrict__ s,
    const float* __restrict__ y0, const float* __restrict__ yc0s,
    const float* __restrict__ y1, const float* __restrict__ yc1s,
    const float* __restrict__ Wcx, const float* __restrict__ Wcy0,
    const float* __restrict__ Wcy1, const float* __restrict__ bcsum,
    const float* __restrict__ ug, const float* __restrict__ hx,
    float* __restrict__ out) {
  constexpr int BS = 2 * 64 + 32;     // 160
  __shared__ float At[64][68];
  __shared__ float Bp[32][BS];
  const int row0 = blockIdx.x * 64;
  const int t    = threadIdx.x;
  const int w    = t >> 5;
  const int lane = t & 31;
  const int row16 = lane & 15;
  const int hk    = lane >> 4;
  const int wr = w >> 1;
  const int wc = w & 1;
  const int colbase = wc * 32;

  v8f acc[2];
  #pragma unroll
  for (int j = 0; j < 2; ++j)
    #pragma unroll
    for (int e = 0; e < 8; ++e) acc[j][e] = 0.0f;

  for (int src = 0; src < 3; ++src) {
    const float* Wsrc = (src == 0) ? Wcx : (src == 1) ? Wcy0 : Wcy1;
    for (int k0 = 0; k0 < FEAT; k0 += 64) {
      const float* Aptr; size_t stride; int off;
      if (src == 0)      { Aptr = (k0 == 0) ? inputs : s;            stride = 64;  off = 0;  }
      else if (src == 1) { if (k0 == 0) { Aptr = y0; stride = 128; off = 0; }
                           else         { Aptr = yc0s; stride = 64; off = 0; } }
      else               { if (k0 == 0) { Aptr = y1; stride = 128; off = 0; }
                           else         { Aptr = yc1s; stride = 64; off = 0; } }
      #pragma unroll
      for (int p = 0; p < 4; ++p) {
        int r  = (t >> 4) + p * 16;
        int c4 = (t & 15) * 4;
        float4 v = *(const float4*)(Aptr + (size_t)(row0 + r) * stride + off + c4);
        At[r][c4 + 0] = v.x; At[r][c4 + 1] = v.y;
        At[r][c4 + 2] = v.z; At[r][c4 + 3] = v.w;
      }
      #pragma unroll
      for (int p = 0; p < 2; ++p) {
        int kp = (t >> 4) + p * 16;
        int o  = (t & 15) * 4;
        const float* s0 = Wsrc + (size_t)(k0 + 2 * kp    ) * UU + o;
        const float* s1 = Wsrc + (size_t)(k0 + 2 * kp + 1) * UU + o;
        st_pair(&Bp[kp][2 * o], *(const float4*)s0, *(const float4*)s1);
      }
      __syncthreads();
      #pragma unroll
      for (int kk = 0; kk < 64; kk += 4) {
        v2f a = *(const v2f*)&At[16 * wr + row16][kk + 2 * hk];
        const int kp = (kk >> 1) + hk;
        #pragma unroll
        for (int j = 0; j < 2; ++j) {
          v2f bv = *(const v2f*)&Bp[kp][2 * (colbase + 16 * j + row16)];
          acc[j] = wmma4(a, bv, acc[j]);
        }
      }
      __syncthreads();
    }
  }

  #pragma unroll
  for (int j = 0; j < 2; ++j) {
    const int col = colbase + 16 * j + row16;
    #pragma unroll
    for (int r = 0; r < 8; ++r) {
      const int gr = row0 + 16 * wr + r + 8 * hk;
      float c = tanhf(acc[j][r] + bcsum[col]);
      float u = ug[(size_t)gr * UU + col];
      float h = hx[(size_t)gr * UU + col];
      out[(size_t)gr * UU + col] = u * h + (1.0f - u) * c;
    }
  }
}

// ---------------------------------------------------------------------------
// Host launcher (graph-capture safe: only kernel launches on `stream`)
// ---------------------------------------------------------------------------
extern "C" void kernel_launch(void* const* d_in, const int* in_sizes, int n_in,
                              void* d_out, int out_size, void* d_ws, size_t ws_size,
                              hipStream_t stream) {
  (void)in_sizes; (void)n_in; (void)out_size; (void)ws_size;
  const float* inputs = (const float*)d_in[0];   // [B, N*U]
  const float* hx     = (const float*)d_in[1];   // [B, N*U]
  const float* adj    = (const float*)d_in[2];   // [B, N, N]
  const float* W0     = (const float*)d_in[3];   // [384, 128]
  const float* b0     = (const float*)d_in[4];
  const float* W1     = (const float*)d_in[5];
  const float* b1     = (const float*)d_in[6];
  const float* Wc0    = (const float*)d_in[7];   // [384, 64]
  const float* bc0    = (const float*)d_in[8];
  const float* Wc1    = (const float*)d_in[9];
  const float* bc1    = (const float*)d_in[10];
  float* out = (float*)d_out;

  // Workspace layout (floats)
  float* p = (float*)d_ws;
  float* d0inv = p; p += NB * NODES;
  float* d1inv = p; p += NB * NODES;
  float* y0    = p; p += (size_t)ROWS * FEAT;    // A0 @ [inputs|hx]
  float* y1    = p; p += (size_t)ROWS * FEAT;    // A1 @ [inputs|hx]
  float* rg    = p; p += (size_t)ROWS * UU;      // reset gate
  float* ug    = p; p += (size_t)ROWS * UU;      // update gate
  float* sbuf  = p; p += (size_t)ROWS * UU;      // r * hx
  float* yc0s  = p; p += (size_t)ROWS * UU;      // A0 @ (r*hx)
  float* yc1s  = p; p += (size_t)ROWS * UU;      // A1 @ (r*hx)
  float* Wx    = p; p += 128 * 128;
  float* Wy0   = p; p += 128 * 128;
  float* Wy1   = p; p += 128 * 128;
  float* Wcx   = p; p += 128 * 64;
  float* Wcy0  = p; p += 128 * 64;
  float* Wcy1  = p; p += 128 * 64;
  float* bsum  = p; p += 128;
  float* bcsum = p; p += 64;

  k_deg_col<<<dim3(NODES / 256, NB), 256, 0, stream>>>(adj, d0inv);
  k_deg_row<<<dim3(NODES / 8,  NB), 256, 0, stream>>>(adj, d1inv);
  k_wcomb<<<128, 128, 0, stream>>>(W0, W1, b0, b1, Wc0, Wc1, bc0, bc1,
                                   Wx, Wy0, Wy1, Wcx, Wcy0, Wcy1, bsum, bcsum);
  k_diffuse<128><<<dim3(NODES / 64, NB, 2), 256, 0, stream>>>(
      adj, inputs, hx, d0inv, d1inv, y0, y1);
  k_gate<<<dim3(ROWS / 64), 256, 0, stream>>>(
      inputs, hx, y0, y1, Wx, Wy0, Wy1, bsum, rg, ug);
  k_rh<<<dim3((ROWS * UU) / 256), 256, 0, stream>>>(rg, hx, sbuf, ROWS * UU);
  k_diffuse<64><<<dim3(NODES / 64, NB, 2), 256, 0, stream>>>(
      adj, sbuf, nullptr, d0inv, d1inv, yc0s, yc1s);
  k_cand<<<dim3(ROWS / 64), 256, 0, stream>>>(
      inputs, sbuf, y0, yc0s, y1, yc1s, Wcx, Wcy0, Wcy1, bcsum, ug, hx, out);
}